// WassersteinProjection_diag_59468117180640
// MI455X (gfx1250) — compile-verified
//
#include <hip/hip_runtime.h>
#include <hip/hip_bf16.h>

typedef __attribute__((ext_vector_type(2))) float v2f;
typedef __attribute__((ext_vector_type(8))) float v8f;

#define ENT_GRID 16
#define ENT_DIM  64
#define EMB_DIM  1024
#define LN_EPS   1e-5f

// One block (256 threads = 8 wave32) per batch row.
// Embedding row staged into LDS with GLOBAL_LOAD_ASYNC_TO_LDS_B128 (ASYNCcnt),
// transform tile warmed with global_prefetch_b8, then each wave runs 8 of the
// 64 d-slices as 4 chained v_wmma_f32_16x16x4_f32 with bias preloaded into C.
__global__ __launch_bounds__(256) void wproj_wmma_kernel(
    const float* __restrict__ ent_emb,      // (B, 1024)
    const int*   __restrict__ proj_ids,     // (B,)
    const float* __restrict__ rel_tran,     // (NREL, 64, 16, 16)
    const float* __restrict__ rel_bias,     // (NREL, 64, 16)
    const float* __restrict__ ln_w,         // (1024,)
    const float* __restrict__ ln_b,         // (1024,)
    float* __restrict__ out)                // (B, 1024)
{
    __shared__ float e_sh[EMB_DIM];   // gathered embedding row
    __shared__ float x_sh[EMB_DIM];   // pre-LN transformed row
    __shared__ float red_s[8];
    __shared__ float red_s2[8];

    const int b    = blockIdx.x;
    const int tid  = threadIdx.x;
    const int wv   = tid >> 5;        // wave id 0..7
    const int lane = tid & 31;
    const int half = lane >> 4;       // 0: lanes 0-15, 1: lanes 16-31
    const int m    = lane & 15;       // matrix row within half

    const int r = proj_ids[b];
    const float* T0 = rel_tran + (size_t)r * (ENT_DIM * ENT_GRID * ENT_GRID);
    const float* B0 = rel_bias + (size_t)r * (ENT_DIM * ENT_GRID);

    // ---- async-stage the embedding row into LDS (4KB: 256 lanes x b128) ----
    {
        unsigned int       lds_off = (unsigned int)(uintptr_t)(&e_sh[0]) + tid * 16u;
        unsigned long long gaddr   =
            (unsigned long long)(uintptr_t)(ent_emb + (size_t)b * EMB_DIM + tid * 4);
        asm volatile("global_load_async_to_lds_b128 %0, %1, off"
                     :
                     : "v"(lds_off), "v"(gaddr)
                     : "memory");
        // Warm the gathered 64KB transform tile (one cacheline per thread).
        __builtin_prefetch(T0 + tid * 64, 0, 1);   // global_prefetch_b8
        asm volatile("s_wait_asynccnt 0x0" ::: "memory");
    }
    __syncthreads();

    // ---- transform phase: y[d, 0:16] = T[d] @ e[d, 0:16] + bias[d, 0:16] ----
    for (int i = 0; i < 8; ++i) {
        const int d = wv * 8 + i;
        const float* T  = T0 + d * (ENT_GRID * ENT_GRID);   // 16x16 row-major [g][h]
        const float* bp = B0 + d * ENT_GRID + half * 8;

        // C/D layout: VGPR v holds M=v (lanes 0-15) and M=v+8 (lanes 16-31),
        // all N columns identical -> preload bias into the accumulator.
        v8f acc;
        {
            const float4 c0 = *(const float4*)(bp);
            const float4 c1 = *(const float4*)(bp + 4);
            acc[0] = c0.x; acc[1] = c0.y; acc[2] = c0.z; acc[3] = c0.w;
            acc[4] = c1.x; acc[5] = c1.y; acc[6] = c1.z; acc[7] = c1.w;
        }

        // K split into 4 chunks of 4. A(16x4): lane m holds row m, VGPR0/1 hold
        // K = {0,1} (lanes 0-15) or K = {2,3} (lanes 16-31) -> contiguous float2.
        // B(4x16): VGPR0/1 hold rows K={0,2}/{1,3} striped across lanes; all
        // columns get the same e value (redundant N, compute is free here).
#pragma unroll
        for (int c4 = 0; c4 < 4; ++c4) {
            const v2f a  = *(const v2f*)(T + m * ENT_GRID + c4 * 4 + half * 2);
            const v2f bv = *(const v2f*)(&e_sh[d * ENT_GRID + c4 * 4 + half * 2]);
            acc = __builtin_amdgcn_wmma_f32_16x16x4_f32(
                /*neg_a=*/false, a, /*neg_b=*/false, bv,
                /*c_mod=*/(short)0, acc, /*reuse_a=*/false, /*reuse_b=*/false);
        }

        // All N columns identical: lane 0 of each half holds y[8*half + 0..7].
        if (m == 0) {
            float4 o0, o1;
            o0.x = acc[0]; o0.y = acc[1]; o0.z = acc[2]; o0.w = acc[3];
            o1.x = acc[4]; o1.y = acc[5]; o1.z = acc[6]; o1.w = acc[7];
            *(float4*)&x_sh[d * ENT_GRID + half * 8]     = o0;
            *(float4*)&x_sh[d * ENT_GRID + half * 8 + 4] = o1;
        }
    }
    __syncthreads();

    // ---- LayerNorm over the 1024-element row ----
    const float4 xv = *(const float4*)&x_sh[tid * 4];
    float s  = xv.x + xv.y + xv.z + xv.w;
    float s2 = xv.x * xv.x + xv.y * xv.y + xv.z * xv.z + xv.w * xv.w;
#pragma unroll
    for (int off = 16; off > 0; off >>= 1) {
        s  += __shfl_down(s,  off, 32);
        s2 += __shfl_down(s2, off, 32);
    }
    if (lane == 0) { red_s[wv] = s; red_s2[wv] = s2; }
    __syncthreads();
    if (tid == 0) {
        float ts = 0.f, t2 = 0.f;
#pragma unroll
        for (int k = 0; k < 8; ++k) { ts += red_s[k]; t2 += red_s2[k]; }
        red_s[0] = ts; red_s2[0] = t2;
    }
    __syncthreads();

    const float mean = red_s[0] * (1.0f / EMB_DIM);
    const float var  = fmaxf(red_s2[0] * (1.0f / EMB_DIM) - mean * mean, 0.0f);
    const float rstd = rsqrtf(var + LN_EPS);

    const float4 w4 = *(const float4*)(ln_w + tid * 4);
    const float4 b4 = *(const float4*)(ln_b + tid * 4);
    float4 o;
    o.x = (xv.x - mean) * rstd * w4.x + b4.x;
    o.y = (xv.y - mean) * rstd * w4.y + b4.y;
    o.z = (xv.z - mean) * rstd * w4.z + b4.z;
    o.w = (xv.w - mean) * rstd * w4.w + b4.w;
    *(float4*)(out + (size_t)b * EMB_DIM + tid * 4) = o;
}

extern "C" void kernel_launch(void* const* d_in, const int* in_sizes, int n_in,
                              void* d_out, int out_size, void* d_ws, size_t ws_size,
                              hipStream_t stream) {
    const float* ent_emb  = (const float*)d_in[0];
    const int*   proj_ids = (const int*)d_in[1];
    const float* rel_tran = (const float*)d_in[2];
    const float* rel_bias = (const float*)d_in[3];
    const float* ln_w     = (const float*)d_in[4];
    const float* ln_b     = (const float*)d_in[5];
    float* out = (float*)d_out;

    const int B = in_sizes[1];   // batch = number of proj ids (8192)
    wproj_wmma_kernel<<<B, 256, 0, stream>>>(
        ent_emb, proj_ids, rel_tran, rel_bias, ln_w, ln_b, out);
}